// ChamferLossSplit_68393059221686
// MI455X (gfx1250) — compile-verified
//
#include <hip/hip_runtime.h>

// CDNA5 / gfx1250 Chamfer split loss.
// dist^2(n,m) = ||x_n||^2 + ||y_m||^2 - 2<x_n,y_m>. Staging A = -2*x makes
// V_WMMA_F32_16X16X4_F32 (K=4 == D) emit c = -2<x,y> directly; one 16x16 tile
// per wmma. Min-reductions run in squared-distance space (sqrt is monotonic)
// with row/col constants hoisted OUT of the reduction:
//   min_m(xx+yy_m+c_m)  = xx + min_m(yy_m+c_m)
//   min_n(xxp_n+yy+c_n) = yy + min_n(xxp_n+c_n)
// Inner loop: 1 wmma + 2 ds_load_b64 + 1 ds_min_i32 + packed adds/min3 only.
// B-fragment loads are software-pipelined one tile ahead to hide LDS latency.

typedef float v2f __attribute__((ext_vector_type(2)));
typedef float v8f __attribute__((ext_vector_type(8)));

#define BATCH 256
#define NPTS  256
#define MPTS  256
#define DDIM  4
#define BIG2F  1.0e20f   // BIG^2 penalty (squared-distance space)
#define HUGEF  3.0e38f   // min identity

__global__ __launch_bounds__(512)
void chamfer_batch_wmma_kernel(const float* __restrict__ target,
                               const float* __restrict__ reco,
                               const int*   __restrict__ in_pid,
                               const int*   __restrict__ out_pid,
                               float*       __restrict__ per_batch)
{
    __shared__ float  sXs[NPTS * DDIM];  // -2 * x rows (A fragments)
    __shared__ float  sY[MPTS * DDIM];   // y rows (B fragments)
    __shared__ float  sXX[NPTS];         // ||x||^2
    __shared__ float  sPenX[NPTS];       // 0 or BIG^2
    __shared__ float2 sYp[MPTS];         // { yy, yy + penY }
    __shared__ float  sMinXY[NPTS];      // min over m of masked d^2
    __shared__ int    sMinYX[MPTS];      // min over n of masked d^2 (float bits)
    __shared__ float  sAcc[6];

    const int b   = blockIdx.x;
    const int tid = threadIdx.x;

    // ---------------- Phase 1: stage + precompute
    if (tid < NPTS) {
        const float4 x = ((const float4*)(target + (size_t)b * NPTS * DDIM))[tid];
        ((float4*)sXs)[tid] = make_float4(-2.f * x.x, -2.f * x.y, -2.f * x.z, -2.f * x.w);
        sXX[tid]   = x.x * x.x + x.y * x.y + x.z * x.z + x.w * x.w;
        sPenX[tid] = (in_pid[(size_t)b * NPTS + tid] != 0) ? 0.0f : BIG2F;
    } else {
        const int t = tid - NPTS;
        const float4 y = ((const float4*)(reco + (size_t)b * MPTS * DDIM))[t];
        ((float4*)sY)[t] = y;
        const float yy   = y.x * y.x + y.y * y.y + y.z * y.z + y.w * y.w;
        const float peny = (out_pid[(size_t)b * MPTS + t] != 0) ? 0.0f : BIG2F;
        sYp[t] = make_float2(yy, yy + peny);
        sMinYX[t] = __float_as_int(HUGEF);
    }
    if (tid < 6) sAcc[tid] = 0.0f;
    __syncthreads();

    // ---------------- Phase 2: WMMA tiles (software-pipelined fragment loads)
    const int wave = tid >> 5;           // n-tile owned by this wave
    const int lane = tid & 31;
    const int half = lane >> 4;          // fp32 frag layout: K pair = 2*half
    const int lm   = lane & 15;
    const int k0   = half * 2;

    // A fragment: row n = 16*wave + lm (already scaled by -2)
    const v2f afrag = *(const v2f*)&sXs[(wave * 16 + lm) * DDIM + k0];

    // C-layout rows of this lane: n = 16*wave + 8*half + v
    float xxr[8], xxp[8], rowmin[8];
#pragma unroll
    for (int v = 0; v < 8; ++v) {
        const int n = wave * 16 + half * 8 + v;
        xxr[v] = sXX[n];
        xxp[v] = xxr[v] + sPenX[n];
        rowmin[v] = HUGEF;               // accumulates min_m (yyp + c)
    }

    // prologue: prefetch tile 0's fragments
    v2f    bnext  = *(const v2f*)&sY[lm * DDIM + k0];
    float2 ypnext = sYp[lm];

#pragma unroll 4
    for (int mt = 0; mt < 16; ++mt) {
        const v2f    bcur  = bnext;
        const float2 ypcur = ypnext;

        // issue next tile's loads BEFORE consuming this tile's fragments
        const int mbn = (((mt + 1) & 15) * 16) + lm;   // wraps on last iter (harmless)
        bnext  = *(const v2f*)&sY[mbn * DDIM + k0];
        ypnext = sYp[mbn];

        v8f c = {};                      // c[v] = -2 * <x_n, y_m>
        c = __builtin_amdgcn_wmma_f32_16x16x4_f32(
                false, afrag, false, bcur, (short)0, c, false, false);

        float colmin = HUGEF;
#pragma unroll
        for (int v = 0; v < 8; ++v) {
            rowmin[v] = fminf(rowmin[v], ypcur.y + c[v]);   // m-masked, xx hoisted
            colmin    = fminf(colmin,    xxp[v] + c[v]);    // n-masked, yy hoisted
        }
        // both halves atomically fold into the same m slot (no cross-lane shuffle)
        atomicMin(&sMinYX[mt * 16 + lm],
                  __float_as_int(colmin + ypcur.x));        // ds_min_i32
    }

    // finish min over m: fold the 16 lanes of each half, then add back xx
#pragma unroll
    for (int v = 0; v < 8; ++v) {
        float r = rowmin[v];
        r = fminf(r, __shfl_xor(r, 1, 32));
        r = fminf(r, __shfl_xor(r, 2, 32));
        r = fminf(r, __shfl_xor(r, 4, 32));
        r = fminf(r, __shfl_xor(r, 8, 32));
        rowmin[v] = xxr[v] + r;
    }
    if (lm == 0) {
#pragma unroll
        for (int v = 0; v < 8; ++v)
            sMinXY[wave * 16 + half * 8 + v] = rowmin[v];
    }
    __syncthreads();

    // ---------------- Phase 3: sqrt of mins + masked sums (ds_add_f32)
    if (tid < NPTS) {
        const int t = tid;
        const bool  mx = (sPenX[t] == 0.0f);
        const float2 yp = sYp[t];
        const bool  my = (yp.y == yp.x);               // penY == 0
        const float dxy = __builtin_amdgcn_sqrtf(fmaxf(sMinXY[t], 0.0f));
        const float dyx = __builtin_amdgcn_sqrtf(fmaxf(__int_as_float(sMinYX[t]), 0.0f));
        atomicAdd(&sAcc[0], mx ? dxy : 0.0f);
        atomicAdd(&sAcc[1], my ? dyx : 0.0f);
        atomicAdd(&sAcc[2], mx ? 1.0f : 0.0f);
        atomicAdd(&sAcc[3], my ? 1.0f : 0.0f);
        atomicAdd(&sAcc[4], mx ? __builtin_amdgcn_sqrtf(sXX[t]) : 0.0f);
        atomicAdd(&sAcc[5], my ? 0.0f : __builtin_amdgcn_sqrtf(yp.x));
    }
    __syncthreads();

    // ---------------- Phase 4: per-batch edge cases
    if (tid == 0) {
        const float sum_xy = sAcc[0], sum_yx = sAcc[1];
        const float fnx = sAcc[2], fny = sAcc[3];
        const float sum_nx_nz = sAcc[4], sum_ny_zero = sAcc[5];
        const float n_in  = fmaxf(1.0f, fnx);
        const float n_out = fmaxf(1.0f, fny);
        const float normal = 0.5f * (sum_xy / n_out + sum_yx / n_in);
        float eucl_nz = normal;
        if (fny == 0.0f)      eucl_nz = sum_nx_nz / n_in;
        else if (fnx == 0.0f) eucl_nz = 0.0f;
        per_batch[b * 2 + 0] = eucl_nz;
        per_batch[b * 2 + 1] = sum_ny_zero / fmaxf(1.0f, (float)MPTS - fny);
    }
}

__global__ __launch_bounds__(256)
void chamfer_finalize_kernel(const float* __restrict__ per_batch,
                             float* __restrict__ out)
{
    __shared__ float s0[BATCH];
    __shared__ float s1[BATCH];
    const int t = threadIdx.x;
    s0[t] = per_batch[t * 2 + 0];
    s1[t] = per_batch[t * 2 + 1];
    __syncthreads();
    for (int off = BATCH / 2; off > 0; off >>= 1) {
        if (t < off) {
            s0[t] += s0[t + off];
            s1[t] += s1[t + off];
        }
        __syncthreads();
    }
    if (t == 0) {
        out[0] = s0[0] * (1.0f / (float)BATCH);
        out[1] = s1[0] * (1.0f / (float)BATCH);
    }
}

extern "C" void kernel_launch(void* const* d_in, const int* in_sizes, int n_in,
                              void* d_out, int out_size, void* d_ws, size_t ws_size,
                              hipStream_t stream)
{
    (void)in_sizes; (void)n_in; (void)out_size; (void)ws_size;
    const float* target  = (const float*)d_in[0];
    const float* reco    = (const float*)d_in[1];
    const int*   in_pid  = (const int*)d_in[2];
    const int*   out_pid = (const int*)d_in[3];
    float* per_batch = (float*)d_ws;     // 256 * 2 floats
    float* out       = (float*)d_out;    // 2 floats

    chamfer_batch_wmma_kernel<<<BATCH, 512, 0, stream>>>(
        target, reco, in_pid, out_pid, per_batch);
    chamfer_finalize_kernel<<<1, 256, 0, stream>>>(per_batch, out);
}